// GNN_77584289235350
// MI455X (gfx1250) — compile-verified
//
#include <hip/hip_runtime.h>
#include <hip/hip_bf16.h>

typedef __attribute__((ext_vector_type(2))) float v2f;
typedef __attribute__((ext_vector_type(8))) float v8f;

#define NNODES 50000
#define NEDGES 800000
#define DDIM   128
#define LDSTRIDE 132                      // 132 mod 64 = 4 -> conflict-free b64 frag loads
#define LDS_BYTES (DDIM * LDSTRIDE * 4)   // 67584 B (dynamic; > 64KB static cap)

// ---------------------------------------------------------------------------
// GEMM: out[M,128] = epilogue( A[M,128] @ W[128,128] + bias )
// EPI: 0 = none, 1 = +bias, 2 = relu(+bias)
// 256 threads = 8 waves; W staged TRANSPOSED (n-major, padded stride) in LDS
// so each B fragment is one contiguous ds_load_b64. Each wave computes a
// 16x128 slab via V_WMMA_F32_16X16X4_F32 (exact fp32 matrix math).
// ---------------------------------------------------------------------------
template <int EPI>
__global__ __launch_bounds__(256) void gnn_gemm128(
    const float* __restrict__ A, const float* __restrict__ W,
    const float* __restrict__ bias, float* __restrict__ out, int M)
{
    extern __shared__ float lwT[];  // [DDIM cols n][LDSTRIDE] transposed weights

    // Stage W transposed: coalesced float4 global reads, scalar DS stores.
    {
        const float4* wv = (const float4*)W;
        const int tid = threadIdx.x;
#pragma unroll
        for (int i = 0; i < 16; ++i) {
            const int vidx = tid + 256 * i;       // float4 index
            const float4 v = wv[vidx];
            const int idx = vidx * 4;             // row-major dword index
            const int k = idx >> 7;               // W row
            const int n = idx & 127;              // W col (n..n+3)
            lwT[(n + 0) * LDSTRIDE + k] = v.x;
            lwT[(n + 1) * LDSTRIDE + k] = v.y;
            lwT[(n + 2) * LDSTRIDE + k] = v.z;
            lwT[(n + 3) * LDSTRIDE + k] = v.w;
        }
    }
    __syncthreads();

    const int wave = threadIdx.x >> 5;   // 0..7
    const int lane = threadIdx.x & 31;
    const int ln   = lane & 15;          // N within tile / M within A frag
    const int koff = (lane >> 4) * 2;    // K sub-offset 0 or 2 (A/B layout)

    const int m0 = blockIdx.x * 128 + wave * 16;
    int rowA = m0 + ln;
    if (rowA >= M) rowA = M - 1;         // clamp loads; stores masked below

    const float* aRow  = A + (size_t)rowA * DDIM + koff;
    const float* bBase = lwT + (size_t)ln * LDSTRIDE + koff;  // + t*16*LDSTRIDE + k0 folds to imm

    v8f acc[8] = {};                     // 8 N-tiles of 16x16 f32

#pragma unroll 4
    for (int k0 = 0; k0 < DDIM; k0 += 4) {
        // A fragment: lane<16 holds A[m0+ln, k0+0..1]; lane>=16 holds K 2..3
        v2f afrag = *(const v2f*)(aRow + k0);
#pragma unroll
        for (int t = 0; t < 8; ++t) {
            // B fragment: contiguous {W[k][n], W[k+1][n]} in transposed LDS
            v2f bfrag = *(const v2f*)(bBase + t * 16 * LDSTRIDE + k0);
            acc[t] = __builtin_amdgcn_wmma_f32_16x16x4_f32(
                false, afrag, false, bfrag, (short)0, acc[t], false, false);
        }
    }

    // Epilogue + store.  D layout: VGPR r -> row (m0 + r) for lanes 0-15,
    // row (m0 + 8 + r) for lanes 16-31; N = t*16 + ln (fixed per lane).
    const int mbase = m0 + (lane >> 4) * 8;
#pragma unroll
    for (int t = 0; t < 8; ++t) {
        const int ncol = t * 16 + ln;
        const float bv = (EPI >= 1) ? bias[ncol] : 0.0f;
#pragma unroll
        for (int r = 0; r < 8; ++r) {
            const int row = mbase + r;
            if (row < M) {
                float v = acc[t][r] + bv;
                if (EPI == 2) v = fmaxf(v, 0.0f);
                out[(size_t)row * DDIM + ncol] = v;
            }
        }
    }
}

// ---------------------------------------------------------------------------
// Edge scatter: acc[dst] += g[src]; cnt[dst]++.  32 lanes per edge, float4
// gather; g and acc are L2-resident (25.6 MB each vs 192 MB L2).
// ---------------------------------------------------------------------------
__global__ __launch_bounds__(256) void gnn_scatter(
    const int* __restrict__ src, const int* __restrict__ dst,
    const float* __restrict__ g, float* __restrict__ acc,
    unsigned* __restrict__ cnt, int nEdges)
{
    int idx = blockIdx.x * 256 + threadIdx.x;
    int e = idx >> 5;
    if (e >= nEdges) return;
    int c = (idx & 31) * 4;
    int s = src[e];
    int d = dst[e];
    const float4 v = *(const float4*)(g + (size_t)s * DDIM + c);
    float* p = acc + (size_t)d * DDIM + c;
    atomicAdd(p + 0, v.x);
    atomicAdd(p + 1, v.y);
    atomicAdd(p + 2, v.z);
    atomicAdd(p + 3, v.w);
    if ((idx & 31) == 0) atomicAdd(cnt + d, 1u);
}

// ---------------------------------------------------------------------------
// Finalize: fold in the self-loop (+g[i], cnt+1), divide by count, add bias.
// ---------------------------------------------------------------------------
__global__ __launch_bounds__(256) void gnn_finalize(
    const float* __restrict__ acc, const float* __restrict__ g,
    const unsigned* __restrict__ cnt, const float* __restrict__ bo,
    float* __restrict__ out, int total)
{
    int idx = blockIdx.x * 256 + threadIdx.x;
    if (idx >= total) return;
    int i = idx >> 7;      // node
    int j = idx & 127;     // channel
    float c = (float)(cnt[i] + 1u);               // self-loop => cnt >= 1
    out[idx] = (acc[idx] + g[idx]) / c + bo[j];
}

// ---------------------------------------------------------------------------
extern "C" void kernel_launch(void* const* d_in, const int* in_sizes, int n_in,
                              void* d_out, int out_size, void* d_ws, size_t ws_size,
                              hipStream_t stream)
{
    (void)in_sizes; (void)n_in; (void)out_size; (void)ws_size;

    const int*   edge_index = (const int*)d_in[0];      // [2, E] (src row, dst row)
    const float* edge_attr  = (const float*)d_in[1];    // [E, 128], only rows 0..N-1 used
    // conv1 weights (d_in[2..7]) are dead code in the reference.
    const float* w1 = (const float*)d_in[8];            // w2_1
    const float* b1 = (const float*)d_in[9];            // b2_1
    const float* w2 = (const float*)d_in[10];           // w2_2
    const float* b2 = (const float*)d_in[11];           // b2_2
    const float* wo = (const float*)d_in[12];           // w2_out
    const float* bo = (const float*)d_in[13];           // b2_out
    float* out = (float*)d_out;

    const size_t rowBytes = (size_t)NNODES * DDIM * sizeof(float);  // 25.6 MB
    char* ws = (char*)d_ws;
    float*    t   = (float*)ws;                   // buffer 0
    float*    h   = (float*)(ws + rowBytes);      // buffer 1
    float*    g   = t;                            // reuse buffer 0 (t dead after GEMM2)
    float*    acc = h;                            // reuse buffer 1 (h dead after GEMM3)
    unsigned* cnt = (unsigned*)(ws + 2 * rowBytes);

    const dim3 blk(256);
    const int gemmBlocks = (NNODES + 127) / 128;  // 391

    // t = relu(edge_attr[0:N] @ w1 + b1)
    gnn_gemm128<2><<<gemmBlocks, blk, LDS_BYTES, stream>>>(edge_attr, w1, b1, t, NNODES);
    // h = t @ w2 + b2
    gnn_gemm128<1><<<gemmBlocks, blk, LDS_BYTES, stream>>>(t, w2, b2, h, NNODES);
    // g = h @ wo   (push output projection before the scatter: row-scaling
    //               by 1/cnt commutes with right-multiplication)
    gnn_gemm128<0><<<gemmBlocks, blk, LDS_BYTES, stream>>>(h, wo, nullptr, g, NNODES);

    hipMemsetAsync(acc, 0, rowBytes, stream);
    hipMemsetAsync(cnt, 0, (size_t)NNODES * sizeof(unsigned), stream);

    // acc[dst] += g[src]; cnt[dst]++  over the 800k real edges
    const int scatterBlocks = (NEDGES * 32) / 256;  // 100000
    gnn_scatter<<<scatterBlocks, blk, 0, stream>>>(
        edge_index, edge_index + NEDGES, g, acc, cnt, NEDGES);

    // out = (acc + g_selfloop) / (cnt+1) + bo
    const int total = NNODES * DDIM;
    gnn_finalize<<<(total + 255) / 256, blk, 0, stream>>>(acc, g, cnt, bo, out, total);
}